// VectorQuantizer_67568425501016
// MI455X (gfx1250) — compile-verified
//
#include <hip/hip_runtime.h>
#include <hip/hip_bf16.h>

#define N_TOK   65536      // 16*4096
#define DIM     64
#define KCODES  1024
#define EMA_DECAY 0.99f
#define COMMIT  0.25f
#define EPS_F   1e-05f

typedef __attribute__((ext_vector_type(2))) float v2f;
typedef __attribute__((ext_vector_type(8))) float v8f;

// ---------------- zero scratch region ----------------
__global__ void vq_zero(float* p, int n) {
    int t = blockIdx.x * blockDim.x + threadIdx.x;
    if (t < n) p[t] = 0.0f;
}

// ---------------- codebook row norms ----------------
__global__ void vq_cnorm(const float* __restrict__ codebook, float* __restrict__ cnorm) {
    int k = blockIdx.x * blockDim.x + threadIdx.x;
    if (k < KCODES) {
        const float* c = codebook + (size_t)k * DIM;
        float s = 0.0f;
        #pragma unroll
        for (int d = 0; d < DIM; ++d) s += c[d] * c[d];
        cnorm[k] = s;
    }
}

// ---- WMMA f32 argmin: one wave = 16-row tile vs all 1024 codes, 2 column tiles in flight ----
__global__ void __launch_bounds__(256)
vq_argmin_wmma(const float* __restrict__ z,
               const float* __restrict__ codebook,
               const float* __restrict__ cnorm,
               int* __restrict__ indices) {
    const int lane    = threadIdx.x & 31;
    const int wave    = threadIdx.x >> 5;
    const int tile    = blockIdx.x * 8 + wave;       // 16-row tile id
    const int row0    = tile * 16;
    const int lane16  = lane & 15;
    const int halfSel = lane >> 4;                   // 0: K=0,1 ; 1: K=2,3
    const int koff    = halfSel * 2;

    // Preload A fragments: a[kk] = z[row0+lane16][4*kk + koff + {0,1}]
    const float* zrow = z + (size_t)(row0 + lane16) * DIM;
    v2f a[16];
    #pragma unroll
    for (int kk = 0; kk < 16; ++kk) {
        a[kk].x = zrow[4 * kk + koff];
        a[kk].y = zrow[4 * kk + koff + 1];
    }

    float best[8];
    int   bidx[8];
    #pragma unroll
    for (int v = 0; v < 8; ++v) { best[v] = 3.4e38f; bidx[v] = 0; }

    for (int c = 0; c < 64; c += 2) {
        const int col0 = c * 16 + lane16;            // this lane's B column (codebook row), tile c
        const int col1 = col0 + 16;                  // tile c+1
        const float* crow0 = codebook + (size_t)col0 * DIM;
        const float* crow1 = codebook + (size_t)col1 * DIM;
        v8f acc0 = {0.f,0.f,0.f,0.f,0.f,0.f,0.f,0.f};
        v8f acc1 = {0.f,0.f,0.f,0.f,0.f,0.f,0.f,0.f};
        #pragma unroll
        for (int kk = 0; kk < 16; ++kk) {
            v2f b0, b1;
            b0.x = crow0[4 * kk + koff];
            b0.y = crow0[4 * kk + koff + 1];
            b1.x = crow1[4 * kk + koff];
            b1.y = crow1[4 * kk + koff + 1];
            // two independent accumulation chains sharing A -> 2x matrix-pipe ILP per wave
            acc0 = __builtin_amdgcn_wmma_f32_16x16x4_f32(
                false, a[kk], false, b0, (short)0, acc0, false, false);
            acc1 = __builtin_amdgcn_wmma_f32_16x16x4_f32(
                false, a[kk], false, b1, (short)0, acc1, false, false);
        }
        const float cn0 = cnorm[col0];
        const float cn1 = cnorm[col1];
        #pragma unroll
        for (int v = 0; v < 8; ++v) {
            // ||c||^2 - 2 z.c  (||z||^2 constant per row); check tile c before c+1
            float d0 = fmaf(-2.0f, acc0[v], cn0);
            float d1 = fmaf(-2.0f, acc1[v], cn1);
            if (d0 < best[v]) { best[v] = d0; bidx[v] = col0; }
            if (d1 < best[v]) { best[v] = d1; bidx[v] = col1; }
        }
    }

    // reduce across the 16 lanes of each half; keep smallest index on ties (argmin semantics)
    #pragma unroll
    for (int v = 0; v < 8; ++v) {
        float bv = best[v];
        int   bi = bidx[v];
        #pragma unroll
        for (int m = 1; m < 16; m <<= 1) {
            float ov = __shfl_xor(bv, m, 32);
            int   oi = __shfl_xor(bi, m, 32);
            if (ov < bv || (ov == bv && oi < bi)) { bv = ov; bi = oi; }
        }
        if (lane16 == 0)
            indices[row0 + v + 8 * halfSel] = bi;    // C-layout: M = v + 8*halfSel
    }
}

// ---------------- gather z_q, commitment loss, cluster scatter ----------------
__global__ void __launch_bounds__(256)
vq_gather(const float* __restrict__ z, const float* __restrict__ codebook,
          const int* __restrict__ indices,
          float* __restrict__ zq_out, float* __restrict__ idx_out_f,
          float* __restrict__ cluster_size, float* __restrict__ cluster_sum,
          float* __restrict__ lossAcc) {
    __shared__ float sred[8];
    const int tid = threadIdx.x;
    const int rowInBlk = tid >> 4;                   // 16 rows per block
    const int l16 = tid & 15;
    const int row = blockIdx.x * 16 + rowInBlk;
    const int idx = indices[row];

    const float4 zv = ((const float4*)(z        + (size_t)row * DIM))[l16];
    const float4 cv = ((const float4*)(codebook + (size_t)idx * DIM))[l16];
    ((float4*)(zq_out + (size_t)row * DIM))[l16] = cv;   // straight-through value == z_q

    float dx = zv.x - cv.x, dy = zv.y - cv.y, dz = zv.z - cv.z, dw = zv.w - cv.w;
    float err = dx*dx + dy*dy + dz*dz + dw*dw;

    float* cs = cluster_sum + (size_t)idx * DIM + l16 * 4;
    atomicAdd(cs + 0, zv.x); atomicAdd(cs + 1, zv.y);
    atomicAdd(cs + 2, zv.z); atomicAdd(cs + 3, zv.w);
    if (l16 == 0) {
        atomicAdd(cluster_size + idx, 1.0f);
        idx_out_f[row] = (float)idx;
    }

    // block reduce err -> single atomic
    #pragma unroll
    for (int m = 16; m >= 1; m >>= 1) err += __shfl_xor(err, m, 32);
    if ((tid & 31) == 0) sred[tid >> 5] = err;
    __syncthreads();
    if (tid < 8) {
        float e = sred[tid];
        #pragma unroll
        for (int m = 4; m >= 1; m >>= 1) e += __shfl_xor(e, m, 32);
        if (tid == 0) atomicAdd(lossAcc, e);
    }
}

// ---------------- EMA blend + n accumulation ----------------
__global__ void vq_ema(const float* __restrict__ ema_cs, const float* __restrict__ ema_w,
                       const float* __restrict__ cluster_size, const float* __restrict__ cluster_sum,
                       float* __restrict__ new_ecs, float* __restrict__ new_ew,
                       float* __restrict__ nAcc) {
    int t = blockIdx.x * blockDim.x + threadIdx.x;   // 65536 = K*D
    int k = t >> 6, d = t & 63;
    float ew = ema_w[t] * EMA_DECAY + (1.0f - EMA_DECAY) * cluster_sum[t];
    new_ew[t] = ew;
    if (d == 0) {
        float ecs = ema_cs[k] * EMA_DECAY + (1.0f - EMA_DECAY) * cluster_size[k];
        new_ecs[k] = ecs;
        atomicAdd(nAcc, ecs);
    }
}

// ---------------- normalize codebook + finalize loss ----------------
__global__ void vq_final(const float* __restrict__ new_ecs, const float* __restrict__ new_ew,
                         const float* __restrict__ nAcc, const float* __restrict__ lossAcc,
                         float* __restrict__ new_codebook, float* __restrict__ vq_loss) {
    int t = blockIdx.x * blockDim.x + threadIdx.x;   // 65536
    int k = t >> 6;
    float n = *nAcc;
    float denom = (new_ecs[k] + EPS_F) / (n + (float)KCODES * EPS_F) * n;
    new_codebook[t] = new_ew[t] / denom;
    if (t == 0)
        *vq_loss = COMMIT * (*lossAcc) / (float)((size_t)N_TOK * DIM);
}

extern "C" void kernel_launch(void* const* d_in, const int* in_sizes, int n_in,
                              void* d_out, int out_size, void* d_ws, size_t ws_size,
                              hipStream_t stream) {
    const float* z        = (const float*)d_in[0];   // [16,4096,64]
    const float* codebook = (const float*)d_in[1];   // [1024,64]
    const float* ema_cs   = (const float*)d_in[2];   // [1024]
    const float* ema_w    = (const float*)d_in[3];   // [1024,64]

    // ----- output layout (flat f32, tuple order) -----
    float* out = (float*)d_out;
    float* zq_out    = out;                                   // 4194304
    float* vq_loss   = out + (size_t)N_TOK * DIM;             // 1
    float* idx_out_f = vq_loss + 1;                           // 65536
    float* new_cb    = idx_out_f + N_TOK;                     // 65536
    float* new_ecs   = new_cb + (size_t)KCODES * DIM;         // 1024
    float* new_ew    = new_ecs + KCODES;                      // 65536

    // ----- workspace layout -----
    float* ws = (float*)d_ws;
    int*   indices      = (int*)ws;                           // 65536 ints
    float* cluster_size = ws + N_TOK;                         // 1024
    float* cluster_sum  = cluster_size + KCODES;              // 65536
    float* lossAcc      = cluster_sum + (size_t)KCODES * DIM; // 1
    float* nAcc         = lossAcc + 1;                        // 1
    float* cnorm        = nAcc + 1;                           // 1024

    // 1) zero accumulators (cluster_size + cluster_sum + loss + n), contiguous
    int nzero = KCODES + KCODES * DIM + 2;
    vq_zero<<<(nzero + 255) / 256, 256, 0, stream>>>(cluster_size, nzero);

    // 2) codebook norms
    vq_cnorm<<<(KCODES + 255) / 256, 256, 0, stream>>>(codebook, cnorm);

    // 3) WMMA f32 distance + argmin (one wave per 16-row tile, 8 waves/block)
    vq_argmin_wmma<<<(N_TOK / 16) / 8, 256, 0, stream>>>(z, codebook, cnorm, indices);

    // 4) gather z_q, loss, cluster scatter (16 rows per 256-thread block)
    vq_gather<<<N_TOK / 16, 256, 0, stream>>>(z, codebook, indices,
                                              zq_out, idx_out_f,
                                              cluster_size, cluster_sum, lossAcc);

    // 5) EMA blend + n = sum(new_ecs)
    vq_ema<<<(KCODES * DIM) / 256, 256, 0, stream>>>(ema_cs, ema_w,
                                                     cluster_size, cluster_sum,
                                                     new_ecs, new_ew, nAcc);

    // 6) normalize codebook + finalize loss
    vq_final<<<(KCODES * DIM) / 256, 256, 0, stream>>>(new_ecs, new_ew, nAcc, lossAcc,
                                                       new_cb, vq_loss);
}